// NeurAutoNet_35940286333268
// MI455X (gfx1250) — compile-verified
//
#include <hip/hip_runtime.h>

typedef float v2f __attribute__((ext_vector_type(2)));
typedef float v8f __attribute__((ext_vector_type(8)));

#define BATCH 32
#define CH    16
#define HH    128
#define WW    128
#define HS    132   // h_lds per-pixel stride in floats (132*4 % 16 == 0, 132%64 spreads banks)

__device__ __forceinline__ float ldx(const float* __restrict__ x, int b, int c, int hh, int ww) {
    if ((unsigned)hh >= (unsigned)HH || (unsigned)ww >= (unsigned)WW) return 0.f;   // zero pad
    return x[((b * CH + c) * HH + hh) * WW + ww];
}

__global__ __launch_bounds__(256) void fused_sobel_mlp_kernel(
    const float* __restrict__ x,  const float* __restrict__ W1,
    const float* __restrict__ b1, const float* __restrict__ W2,
    const float* __restrict__ b2, const float* __restrict__ u,
    float* __restrict__ out, float* __restrict__ ch3ws)
{
    __shared__ float h_lds[8 * 16 * HS];          // 67.5 KB, wave-private slices

    const int bh   = blockIdx.x;                  // b*HH + h
    const int b    = bh >> 7;
    const int h    = bh & (HH - 1);
    const int wave = threadIdx.x >> 5;
    const int lane = threadIdx.x & 31;
    const int p    = lane & 15;                   // pixel within wave tile (N index)
    const int half = lane >> 4;
    const int w    = wave * 16 + p;               // pixel column, 8 waves cover the row
    float* hw = &h_lds[wave * 16 * HS];

    // ---------------- Sobel features, kept in-lane ----------------
    // lane (p, half) computes channels c with (c mod 4) in {2*half, 2*half+1}
    v2f gx2[4], gy2[4], xc2[4];
    #pragma unroll
    for (int i = 0; i < 4; ++i) {
        #pragma unroll
        for (int j = 0; j < 2; ++j) {
            const int c = 4 * i + 2 * half + j;
            const float tl = ldx(x,b,c,h-1,w-1), tm = ldx(x,b,c,h-1,w), tr = ldx(x,b,c,h-1,w+1);
            const float ml = ldx(x,b,c,h  ,w-1), mm = ldx(x,b,c,h  ,w), mr = ldx(x,b,c,h  ,w+1);
            const float bl = ldx(x,b,c,h+1,w-1), bm = ldx(x,b,c,h+1,w), br = ldx(x,b,c,h+1,w+1);
            gx2[i][j] = (tr - tl) + 2.f * (mr - ml) + (br - bl);
            gy2[i][j] = (bl + 2.f * bm + br) - (tl + 2.f * tm + tr);
            xc2[i][j] = mm;
        }
    }

    // ---------------- layer 1: h = relu(W1 @ f + b1) ----------------
    // 8 M-tiles of 16, K = 48 = 12 steps of 4 (f32 WMMA, exact f32 math)
    v8f acc[8] = {};
    #pragma unroll
    for (int kk = 0; kk < 12; ++kk) {
        const v2f bv = (kk < 4) ? gx2[kk] : (kk < 8) ? gy2[kk - 4] : xc2[kk - 8];
        #pragma unroll
        for (int m = 0; m < 8; ++m) {
            // A-matrix lane layout: lane p holds W1[16m+p][4kk + 2*half + {0,1}]
            const v2f av = *(const v2f*)&W1[(16 * m + p) * 48 + 4 * kk + 2 * half];
            acc[m] = __builtin_amdgcn_wmma_f32_16x16x4_f32(
                false, av, false, bv, (short)0, acc[m], false, false);
        }
    }

    // bias + relu, scatter to LDS in [pixel][feature] layout for the K-reshuffle
    #pragma unroll
    for (int m = 0; m < 8; ++m) {
        // D layout: vgpr r, lane (p,half) holds h[o = 16m + r + 8*half][p]
        const float4 bA = *(const float4*)&b1[16 * m + 8 * half];
        const float4 bB = *(const float4*)&b1[16 * m + 8 * half + 4];
        float4 hA, hB;
        hA.x = fmaxf(acc[m][0] + bA.x, 0.f);
        hA.y = fmaxf(acc[m][1] + bA.y, 0.f);
        hA.z = fmaxf(acc[m][2] + bA.z, 0.f);
        hA.w = fmaxf(acc[m][3] + bA.w, 0.f);
        hB.x = fmaxf(acc[m][4] + bB.x, 0.f);
        hB.y = fmaxf(acc[m][5] + bB.y, 0.f);
        hB.z = fmaxf(acc[m][6] + bB.z, 0.f);
        hB.w = fmaxf(acc[m][7] + bB.w, 0.f);
        const int base = p * HS + 16 * m + 8 * half;
        *(float4*)&hw[base]     = hA;
        *(float4*)&hw[base + 4] = hB;
    }
    // wave-private LDS region; DS ops from one wave are processed in order -> no barrier

    // ---------------- layer 2: y = W2 @ h + b2 (M=16 ch, K=128) ----------------
    v8f acc2 = {};
    #pragma unroll
    for (int kk = 0; kk < 32; ++kk) {
        const v2f bv = *(const v2f*)&hw[p * HS + 4 * kk + 2 * half];      // h[k..k+1][p]
        const v2f av = *(const v2f*)&W2[p * 128 + 4 * kk + 2 * half];     // W2[c=p][k..k+1]
        acc2 = __builtin_amdgcn_wmma_f32_16x16x4_f32(
            false, av, false, bv, (short)0, acc2, false, false);
    }

    // ---------------- epilogue: bias + dropout mask + residual ----------------
    #pragma unroll
    for (int r = 0; r < 8; ++r) {
        const int c = 8 * half + r;                       // D layout: row = r + 8*half
        const float y = acc2[r] + b2[c];
        const int idx = ((b * CH + c) * HH + h) * WW + w;
        const float mask = (u[idx] >= 0.2f) ? 1.25f : 0.f;   // (u>=p)/(1-p), p=0.2
        const float xn = x[idx] + y * mask;
        out[idx] = xn;
        if (c == 3) ch3ws[(b * HH + h) * WW + w] = xn;    // unpooled ch3 copy for pass 2
    }
}

// 3x3 max-pool on channel 3 (needs cross-block neighbors -> separate pass)
__global__ __launch_bounds__(256) void pool_gate_kernel(
    const float* __restrict__ ch3ws, float* __restrict__ out)
{
    const int idx = blockIdx.x * 256 + threadIdx.x;       // over B*H*W
    const int w = idx & (WW - 1);
    const int h = (idx >> 7) & (HH - 1);
    const int b = idx >> 14;
    const float* plane = ch3ws + b * HH * WW;
    const float center = plane[h * WW + w];
    float m = -3.402823466e38f;                           // padding acts as -inf
    #pragma unroll
    for (int dy = -1; dy <= 1; ++dy) {
        #pragma unroll
        for (int dx = -1; dx <= 1; ++dx) {
            const int hh = h + dy, ww = w + dx;
            if ((unsigned)hh < (unsigned)HH && (unsigned)ww < (unsigned)WW)
                m = fmaxf(m, plane[hh * WW + ww]);
        }
    }
    out[((b * CH + 3) * HH + h) * WW + w] = (m > 0.1f) ? center : 0.f;
}

extern "C" void kernel_launch(void* const* d_in, const int* in_sizes, int n_in,
                              void* d_out, int out_size, void* d_ws, size_t ws_size,
                              hipStream_t stream) {
    (void)in_sizes; (void)n_in; (void)out_size; (void)ws_size;
    const float* x  = (const float*)d_in[0];
    const float* W1 = (const float*)d_in[1];
    const float* b1 = (const float*)d_in[2];
    const float* W2 = (const float*)d_in[3];
    const float* b2 = (const float*)d_in[4];
    const float* u  = (const float*)d_in[5];
    float* out = (float*)d_out;
    float* ws  = (float*)d_ws;   // B*H*W floats = 2 MB ch3 staging

    fused_sobel_mlp_kernel<<<BATCH * HH, 256, 0, stream>>>(x, W1, b1, W2, b2, u, out, ws);
    pool_gate_kernel<<<(BATCH * HH * WW) / 256, 256, 0, stream>>>(ws, out);
}